// project_C_shape_simple_12610023981118
// MI455X (gfx1250) — compile-verified
//
#include <hip/hip_runtime.h>
#include <hip/hip_bf16.h>

typedef float v2f __attribute__((ext_vector_type(2)));
typedef float v8f __attribute__((ext_vector_type(8)));

struct F3 { float x, y, z; };

// One wave (32 lanes) handles 2 constraints (16 particles each, 1 per lane).
// Mass-weighted sums (m*x, m*y, m*z, m) per constraint are reduced with
// shfl_xor presums (groups of 4) + one V_WMMA_F32_16X16X4_F32 row-sum
// (A = partial sums in documented 16x4 layout, B = all-ones so B's VGPR
// layout cannot affect correctness). Row r = c_half*8 + q so the documented
// C/D layout delivers each half-wave its own constraint's totals in D[0..3]
// with no post-WMMA broadcast.
__global__ __launch_bounds__(256, 4) void project_shape_wmma_kernel(
    const float* __restrict__ V_predict,
    const float* __restrict__ L_last,
    const float* __restrict__ V_w,
    const float* __restrict__ V_mass,
    const int*   __restrict__ C_shape,
    const float* __restrict__ C_init,
    const float* __restrict__ V_comp,
    float* __restrict__ out,
    int C_total, int N)
{
    const int lane = threadIdx.x & 31;
    const int wid  = blockIdx.x * (blockDim.x >> 5) + (threadIdx.x >> 5);

    // Wave-uniform guard: keeps EXEC all-ones for surviving waves (WMMA req).
    if (wid * 2 >= C_total) return;

    int c = wid * 2 + (lane >> 4);
    if (c >= C_total) c = C_total - 1;   // odd-C tail: benign duplicate work

    const int p    = lane & 15;
    const int slot = c * 16 + p;
    const int v    = C_shape[slot];

    const F3 P = ((const F3*)V_predict)[v];   // global_load_b96 gather
    const F3 I = ((const F3*)C_init)[slot];
    const float m    = V_mass[v];
    const float w    = V_w[v];
    const float comp = V_comp[v];

    // Per-lane quantities to reduce over the 16 particles of this constraint.
    float r0 = m * P.x, r1 = m * P.y, r2 = m * P.z, r3 = m;

    // Butterfly presum within aligned groups of 4 lanes -> group sums g0..g3
    // per constraint half (lanes c*16 + 4g .. +3 all hold groupsum(c,*,g)).
    r0 += __shfl_xor(r0, 1, 32); r1 += __shfl_xor(r1, 1, 32);
    r2 += __shfl_xor(r2, 1, 32); r3 += __shfl_xor(r3, 1, 32);
    r0 += __shfl_xor(r0, 2, 32); r1 += __shfl_xor(r1, 2, 32);
    r2 += __shfl_xor(r2, 2, 32); r3 += __shfl_xor(r3, 2, 32);

    // Marshal into WMMA A (16x4 f32). Target row r = cL*8 + q:
    //   rows 0..3  = constraint half 0, q = 0..3 (m*x, m*y, m*z, m)
    //   rows 8..11 = constraint half 1, q = 0..3
    // A layout (documented): lane l<16 -> row l, VGPR0=K0, VGPR1=K1;
    //                        lane l>=16 -> row l-16, VGPR0=K2, VGPR1=K3.
    const int  rl  = lane & 15;
    const int  hi  = lane >> 4;
    const int  cL  = rl >> 3;           // which constraint half this row is
    const int  q   = rl & 3;            // which quantity this row carries
    const bool act = (rl & 7) < 4;      // rows 4..7 / 12..15 are zero
    const int  s0  = cL * 16 + 8 * hi;  // source lane for K0 (hi=0) / K2 (hi=1)
    const int  s1  = s0 + 4;            // source lane for K1 / K3

    float a00 = __shfl(r0, s0, 32), a01 = __shfl(r1, s0, 32);
    float a02 = __shfl(r2, s0, 32), a03 = __shfl(r3, s0, 32);
    float a10 = __shfl(r0, s1, 32), a11 = __shfl(r1, s1, 32);
    float a12 = __shfl(r2, s1, 32), a13 = __shfl(r3, s1, 32);

    float A0 = (q == 0) ? a00 : (q == 1) ? a01 : (q == 2) ? a02 : a03;
    float A1 = (q == 0) ? a10 : (q == 1) ? a11 : (q == 2) ? a12 : a13;
    A0 = act ? A0 : 0.0f;
    A1 = act ? A1 : 0.0f;

    v2f a; a[0] = A0;   a[1] = A1;
    v2f b; b[0] = 1.0f; b[1] = 1.0f;    // all-ones B: D[m][n] = sum_k A[m][k]
    v8f acc = {0.f, 0.f, 0.f, 0.f, 0.f, 0.f, 0.f, 0.f};

    acc = __builtin_amdgcn_wmma_f32_16x16x4_f32(
        /*neg_a=*/false, a, /*neg_b=*/false, b,
        /*c_mod=*/(short)0, acc, /*reuse_a=*/false, /*reuse_b=*/false);

    // D layout: lanes 0-15 read rows 0-7, lanes 16-31 read rows 8-15.
    // VGPR j = row (hi*8 + j) = totals of this lane's own constraint, q=j.
    const float Sm   = acc[3];
    const float inv  = 1.0f / Sm;
    const float comx = acc[0] * inv;
    const float comy = acc[1] * inv;
    const float comz = acc[2] * inv;

    // rot = Vh^T * Vh = I (reference discards U,s; det(V*Vh)=1), so
    // delta_x = init - local_pred;  delta_pos = (w/comp) * delta_x.
    const float k  = w / comp;
    const float lx = P.x - comx, ly = P.y - comy, lz = P.z - comz;

    const float ox = P.x + k * (I.x - lx);
    const float oy = P.y + k * (I.y - ly);
    const float oz = P.z + k * (I.z - lz);

    // Disjoint-partition scatter: plain write, no atomics.
    out[3 * v + 0] = ox;
    out[3 * v + 1] = oy;
    out[3 * v + 2] = oz;

    // Second tuple output: L_last passthrough (one lane per constraint).
    if (p == 0) out[3 * N + c] = L_last[c];
}

extern "C" void kernel_launch(void* const* d_in, const int* in_sizes, int n_in,
                              void* d_out, int out_size, void* d_ws, size_t ws_size,
                              hipStream_t stream) {
    const float* V_predict = (const float*)d_in[0];   // [N,3]
    const float* L_last    = (const float*)d_in[1];   // [C,1]
    const float* V_w       = (const float*)d_in[2];   // [N,1]
    const float* V_mass    = (const float*)d_in[3];   // [N,1]
    const int*   C_shape   = (const int*)  d_in[4];   // [C,16]
    const float* C_init    = (const float*)d_in[5];   // [C,16,3]
    const float* V_comp    = (const float*)d_in[6];   // [N,1]
    float* out = (float*)d_out;                       // [N*3 | C] concat

    const int C_total = in_sizes[1];   // L_last element count = C
    const int N       = in_sizes[2];   // V_w element count = N

    const int waves  = (C_total + 1) / 2;       // 2 constraints per wave
    const int blocks = (waves + 7) / 8;         // 8 waves per 256-thread block

    project_shape_wmma_kernel<<<blocks, 256, 0, stream>>>(
        V_predict, L_last, V_w, V_mass, C_shape, C_init, V_comp,
        out, C_total, N);
}